// GCNMODEL_66451734003819
// MI455X (gfx1250) — compile-verified
//
#include <hip/hip_runtime.h>

typedef float v2f __attribute__((ext_vector_type(2)));
typedef float v4f __attribute__((ext_vector_type(4)));
typedef float v8f __attribute__((ext_vector_type(8)));

#define NNODES 50000
#define NEDGES 800000
#define DIM    512
#define HID    512
#define NLAYERS 3
#define BNEPS  1e-5f
#define NBLK   196          // 196*256 = 50176 >= NNODES (scan/elementwise grids)

// ================= CSR build (per-target adjacency) =================
__global__ void cnt_zero(int* cnt) {
    int i = blockIdx.x * 256 + threadIdx.x;
    if (i < NNODES) cnt[i] = 0;
}
__global__ void cnt_count(const int* __restrict__ cols, int* cnt) {
    int e = blockIdx.x * blockDim.x + threadIdx.x;
    if (e < NEDGES) atomicAdd(&cnt[cols[e]], 1);
}
// dinv[i] = rsqrt(indeg + 1)  (self loop contributes 1; always > 0)
__global__ void make_dinv(const int* __restrict__ cnt, float* dinv) {
    int i = blockIdx.x * 256 + threadIdx.x;
    if (i < NNODES) dinv[i] = rsqrtf((float)cnt[i] + 1.0f);
}

// block-local exclusive scan (Hillis-Steele in LDS), emits block totals
__global__ __launch_bounds__(256) void scan_block(const int* __restrict__ cnt,
                                                  int* rowptr, int* bsum) {
    __shared__ int sdata[256];
    int tid = threadIdx.x;
    int i = blockIdx.x * 256 + tid;
    int v = (i < NNODES) ? cnt[i] : 0;
    sdata[tid] = v;
    __syncthreads();
    for (int off = 1; off < 256; off <<= 1) {
        int tv = 0;
        if (tid >= off) tv = sdata[tid - off];
        __syncthreads();
        sdata[tid] += tv;
        __syncthreads();
    }
    if (i < NNODES) rowptr[i] = sdata[tid] - v;   // exclusive within block
    if (tid == 255) bsum[blockIdx.x] = sdata[255];
}
__global__ void scan_tops(const int* __restrict__ bsum, int* bsumoff, int* rowptr) {
    if (threadIdx.x == 0 && blockIdx.x == 0) {
        int run = 0;
        for (int b = 0; b < NBLK; ++b) { bsumoff[b] = run; run += bsum[b]; }
        rowptr[NNODES] = run;                     // == NEDGES
    }
}
__global__ void scan_add(int* rowptr, const int* __restrict__ bsumoff) {
    int i = blockIdx.x * 256 + threadIdx.x;
    if (i < NNODES) rowptr[i] += bsumoff[blockIdx.x];
}
__global__ void cursor_init(const int* __restrict__ rowptr, int* cursor) {
    int i = blockIdx.x * 256 + threadIdx.x;
    if (i < NNODES) cursor[i] = rowptr[i];
}
__global__ void csr_fill(const int* __restrict__ rows, const int* __restrict__ cols,
                         int* cursor, int* srcidx) {
    int e = blockIdx.x * blockDim.x + threadIdx.x;
    if (e < NEDGES) {
        int c = cols[e];
        int pos = atomicAdd(&cursor[c], 1);
        srcidx[pos] = rows[e];
    }
}

// ============ GEMM: T = A(N,512) x W(512,512) + bias, fp32 WMMA ============
// 64-row strips: block = 256 = 8 waves; wave w -> cols [w*64,w*64+64) for 4 row
// tiles; B fragment reused across the 4 row tiles (4x less B L2 traffic).
// Ragged last block: clamp A-row loads (EXEC stays all-1s), predicate stores.
__global__ __launch_bounds__(256) void gemm_wmma_f32(
    const float* __restrict__ A, const float* __restrict__ W,
    const float* __restrict__ bias, float* __restrict__ T)
{
    const int lane = threadIdx.x & 31;
    const int wave = threadIdx.x >> 5;
    const int half = lane >> 4;     // 0: K pair k..k+1, 1: k+2..k+3
    const int lm   = lane & 15;
    const int row0 = blockIdx.x * 64;
    const int colbase = wave * 64;

    const float* Arow[4];
    #pragma unroll
    for (int mt = 0; mt < 4; ++mt) {
        int rr = row0 + mt * 16 + lm;
        if (rr > NNODES - 1) rr = NNODES - 1;     // clamp for ragged tail
        Arow[mt] = A + (size_t)rr * DIM;
    }

    v8f acc[4][4];
    #pragma unroll
    for (int ct = 0; ct < 4; ++ct) {
        float bv = bias[colbase + ct * 16 + lm];  // bias depends only on column
        #pragma unroll
        for (int mt = 0; mt < 4; ++mt)
            #pragma unroll
            for (int r = 0; r < 8; ++r) acc[mt][ct][r] = bv;
    }

    for (int k = 0; k < DIM; k += 4) {
        v2f a[4];
        #pragma unroll
        for (int mt = 0; mt < 4; ++mt) {
            const float* ap = Arow[mt] + k + 2 * half;
            a[mt].x = ap[0]; a[mt].y = ap[1];
        }
        #pragma unroll
        for (int ct = 0; ct < 4; ++ct) {
            const int c = colbase + ct * 16 + lm;
            v2f b;
            b.x = W[(size_t)(k + 2 * half)     * HID + c];
            b.y = W[(size_t)(k + 2 * half + 1) * HID + c];
            #pragma unroll
            for (int mt = 0; mt < 4; ++mt)
                acc[mt][ct] = __builtin_amdgcn_wmma_f32_16x16x4_f32(
                    false, a[mt], false, b, (short)0, acc[mt][ct], false, false);
        }
    }

    // C/D layout: VGPR r, lane -> (M = half*8 + r, N = lm)
    #pragma unroll
    for (int mt = 0; mt < 4; ++mt) {
        #pragma unroll
        for (int r = 0; r < 8; ++r) {
            int row = row0 + mt * 16 + half * 8 + r;
            if (row < NNODES) {
                #pragma unroll
                for (int ct = 0; ct < 4; ++ct)
                    T[(size_t)row * HID + colbase + ct * 16 + lm] = acc[mt][ct][r];
            }
        }
    }
}

// ============ pull-based aggregation: agg[n] = sum_{s in in(n)} w*t[s] ============
// one block per destination node; 128 threads x float4 = 512 features.
// t (102MB) stays resident in the 192MB L2, so gathers are L2 hits; output is a
// single plain store per element (no atomics).
__global__ __launch_bounds__(128) void gather_agg(
    const float* __restrict__ t, const float* __restrict__ dinv,
    const int* __restrict__ rowptr, const int* __restrict__ srcidx,
    float* __restrict__ agg)
{
    int n = blockIdx.x;
    int f = threadIdx.x * 4;
    float di = dinv[n];
    float wself = di * di;                         // self-loop weight
    v4f v = *(const v4f*)(t + (size_t)n * HID + f);
    v4f acc;
    acc.x = v.x * wself; acc.y = v.y * wself; acc.z = v.z * wself; acc.w = v.w * wself;

    int j0 = rowptr[n], j1 = rowptr[n + 1];
    for (int j = j0; j < j1; ++j) {
        int s = srcidx[j];
        float w = di * dinv[s];
        v4f u = *(const v4f*)(t + (size_t)s * HID + f);
        acc.x += u.x * w; acc.y += u.y * w; acc.z += u.z * w; acc.w += u.w * w;
    }
    *(v4f*)(agg + (size_t)n * HID + f) = acc;
}

// ================= batch norm =================
__global__ void bn_zero(float* colsum, float* colsumsq) {
    int i = blockIdx.x * blockDim.x + threadIdx.x;
    if (i < HID) { colsum[i] = 0.0f; colsumsq[i] = 0.0f; }
}
// grid (HID/256, 512): each thread reduces a 98-row stripe of one column
__global__ __launch_bounds__(256) void bn_stats(
    const float* __restrict__ agg, float* colsum, float* colsumsq)
{
    int c = blockIdx.x * 256 + threadIdx.x;
    int n0 = blockIdx.y * 98;
    int n1 = min(n0 + 98, NNODES);
    float s = 0.0f, sq = 0.0f;
    for (int n = n0; n < n1; ++n) {
        float v = agg[(size_t)n * HID + c];
        s += v; sq += v * v;
    }
    atomicAdd(&colsum[c], s);
    atomicAdd(&colsumsq[c], sq);
}
__global__ void bn_apply_relu(const float* __restrict__ agg,
                              const float* __restrict__ colsum,
                              const float* __restrict__ colsumsq,
                              const float* __restrict__ gamma,
                              const float* __restrict__ beta,
                              float* __restrict__ out)
{
    int i = blockIdx.x * blockDim.x + threadIdx.x;
    if (i < NNODES * HID) {
        int c = i & (HID - 1);
        float mean = colsum[c] * (1.0f / NNODES);
        float var  = colsumsq[c] * (1.0f / NNODES) - mean * mean;
        float g = gamma[c] * rsqrtf(var + BNEPS);
        float v = (agg[i] - mean) * g + beta[c];
        out[i] = fmaxf(v, 0.0f);
    }
}
__global__ void dup_out(const float* __restrict__ src, float* __restrict__ dst) {
    int i = blockIdx.x * blockDim.x + threadIdx.x;
    if (i < NNODES * HID) dst[i] = src[i];
}

// ================= launch =================
extern "C" void kernel_launch(void* const* d_in, const int* in_sizes, int n_in,
                              void* d_out, int out_size, void* d_ws, size_t ws_size,
                              hipStream_t stream)
{
    const float* x     = (const float*)d_in[0];       // [N, D]
    const int*   ei    = (const int*)d_in[1];         // [2, E] int32
    const float* Wall  = (const float*)d_in[2];       // [L, D, H]
    const float* ball  = (const float*)d_in[3];       // [L, H]
    const float* gamma = (const float*)d_in[4];       // [H]
    const float* beta  = (const float*)d_in[5];       // [H]

    float* out = (float*)d_out;
    float* h   = out;                                 // first N*H: current features
    float* agg = out + (size_t)NNODES * HID;          // second N*H: aggregation buffer

    float* t        = (float*)d_ws;                   // N*H floats
    float* dinv     = t + (size_t)NNODES * HID;       // N
    float* colsum   = dinv + NNODES;                  // H
    float* colsumsq = colsum + HID;                   // H
    int*   cnt      = (int*)(colsumsq + HID);         // N
    int*   rowptr   = cnt + NNODES;                   // N+1
    int*   cursor   = rowptr + NNODES + 1;            // N
    int*   srcidx   = cursor + NNODES;                // E
    int*   bsum     = srcidx + NEDGES;                // NBLK
    int*   bsumoff  = bsum + 256;                     // NBLK

    const int* rows = ei;                             // edge_index[0] (sources)
    const int* cols = ei + NEDGES;                    // edge_index[1] (targets)

    // ---- degrees + CSR (rebuilt every call; no cross-call state) ----
    cnt_zero   <<<NBLK, 256, 0, stream>>>(cnt);
    cnt_count  <<<(NEDGES + 255) / 256, 256, 0, stream>>>(cols, cnt);
    make_dinv  <<<NBLK, 256, 0, stream>>>(cnt, dinv);
    scan_block <<<NBLK, 256, 0, stream>>>(cnt, rowptr, bsum);
    scan_tops  <<<1, 1, 0, stream>>>(bsum, bsumoff, rowptr);
    scan_add   <<<NBLK, 256, 0, stream>>>(rowptr, bsumoff);
    cursor_init<<<NBLK, 256, 0, stream>>>(rowptr, cursor);
    csr_fill   <<<(NEDGES + 255) / 256, 256, 0, stream>>>(rows, cols, cursor, srcidx);

    // ---- layers ----
    for (int l = 0; l < NLAYERS; ++l) {
        const float* hin = (l == 0) ? x : h;
        gemm_wmma_f32<<<(NNODES + 63) / 64, 256, 0, stream>>>(
            hin, Wall + (size_t)l * DIM * HID, ball + (size_t)l * HID, t);
        gather_agg<<<NNODES, 128, 0, stream>>>(t, dinv, rowptr, srcidx, agg);
        bn_zero<<<(HID + 255) / 256, 256, 0, stream>>>(colsum, colsumsq);
        dim3 sg(HID / 256, 512);
        bn_stats<<<sg, 256, 0, stream>>>(agg, colsum, colsumsq);
        bn_apply_relu<<<(NNODES * HID + 255) / 256, 256, 0, stream>>>(
            agg, colsum, colsumsq, gamma, beta, h);
    }
    dup_out<<<(NNODES * HID + 255) / 256, 256, 0, stream>>>(h, agg);
}